// nUIV_rhs_8134668058827
// MI455X (gfx1250) — compile-verified
//
#include <hip/hip_runtime.h>
#include <hip/hip_bf16.h>

// N = 8192 nUIV RHS. Dominant cost: coupling = ts.T @ V  (268 MB of fp32 ts,
// streamed once -> memory bound; floor ~11.5us at 23.3 TB/s).
// Strategy:
//   k0: extract V (stride-3 gather from state) into ws.
//   k1: WMMA f32 16x16x4 mat-vec partials: grid (N/16 j-tiles, SPLIT i-slices),
//       1 wave per block. A-frag = ts^T tile (coalesced 16-wide j loads, NT
//       temporal hint since ts has zero reuse), B-frag = V broadcast -> every
//       column of D is the partial coupling. Base pointer hoisted so all loads
//       use immediate offsets; one pointer bump per 16 rows.
//   k2: reduce slices, subtract diag(ts)*V, elementwise U/I/V math, write
//       interleaved [dU,dI,dV] output.

#define NN 8192

typedef __attribute__((ext_vector_type(2))) float v2f;
typedef __attribute__((ext_vector_type(8))) float v8f;

__global__ void __launch_bounds__(256)
extract_V_kernel(const float* __restrict__ state, float* __restrict__ Vvec) {
    int i = blockIdx.x * blockDim.x + threadIdx.x;
    if (i < NN) Vvec[i] = state[3 * i + 2];
}

// One wave (32 threads) per block. blockIdx.x = j-tile (16 cols),
// blockIdx.y = i-slice. rows_per_slice = NN / gridDim.y (multiple of 16).
__global__ void __launch_bounds__(32)
matvec_partial_kernel(const float* __restrict__ ts,
                      const float* __restrict__ Vvec,
                      float* __restrict__ partials) {
    const int lane  = threadIdx.x;       // 0..31
    const int half  = lane >> 4;         // 0: K=0,1  1: K=2,3 (A-frag layout)
    const int m     = lane & 15;         // M index (the j within the tile)
    const int jbase = blockIdx.x * 16;
    const int rows_per_slice = NN / gridDim.y;
    const int i0 = blockIdx.y * rows_per_slice;

    // Per-lane column pointer, pre-offset by this half's starting K rows:
    // A[m][k] = ts[i0 + k][jbase + m],  half 0 covers k=0,1  half 1 covers k=2,3
    const float* __restrict__ ph =
        ts + (size_t)(i0 + 2 * half) * NN + (size_t)jbase + (size_t)m;
    const float* __restrict__ vph = Vvec + i0 + 2 * half;

    v8f acc0 = {};
    v8f acc1 = {};
    const int steps = rows_per_slice / 16;
    for (int s = 0; s < steps; ++s) {
#pragma unroll
        for (int u = 0; u < 4; ++u) {          // rows u*4 .. u*4+3 of this step
            v2f a, b;
            a.x = __builtin_nontemporal_load(ph + (size_t)(u * 4) * NN);
            a.y = __builtin_nontemporal_load(ph + (size_t)(u * 4 + 1) * NN);
            b.x = vph[u * 4];
            b.y = vph[u * 4 + 1];
            if (u & 1)
                acc1 = __builtin_amdgcn_wmma_f32_16x16x4_f32(
                    false, a, false, b, (short)0, acc1, false, false);
            else
                acc0 = __builtin_amdgcn_wmma_f32_16x16x4_f32(
                    false, a, false, b, (short)0, acc0, false, false);
        }
        ph  += (size_t)16 * NN;
        vph += 16;
    }
    v8f acc = acc0 + acc1;

    // D[m][n] identical across n. Column n=0:
    //   lane 0  holds M=0..7  in vgpr 0..7
    //   lane 16 holds M=8..15 in vgpr 0..7
    if (m == 0) {
        float* __restrict__ dst =
            partials + (size_t)blockIdx.y * NN + (size_t)jbase + half * 8;
#pragma unroll
        for (int e = 0; e < 8; ++e) dst[e] = acc[e];
    }
}

__global__ void __launch_bounds__(256)
finalize_kernel(const float* __restrict__ state,
                const float* __restrict__ betas,
                const float* __restrict__ deltas,
                const float* __restrict__ cs,
                const float* __restrict__ ps,
                const float* __restrict__ ts,
                const float* __restrict__ partials,
                float* __restrict__ out,
                int split) {
    int j = blockIdx.x * blockDim.x + threadIdx.x;
    if (j >= NN) return;

    float U = state[3 * j + 0];
    float I = state[3 * j + 1];
    float V = state[3 * j + 2];

    float beta  = betas[j]  * betas[j];
    float delta = deltas[j] * deltas[j];
    float c     = cs[j]     * cs[j];
    float p     = ps[j]     * ps[j];

    float coupling = 0.0f;
    for (int s = 0; s < split; ++s)
        coupling += partials[(size_t)s * NN + j];
    coupling -= ts[(size_t)j * NN + j] * V;

    float infect = beta * U * V;
    out[3 * j + 0] = -infect;
    out[3 * j + 1] = infect - delta * I;
    out[3 * j + 2] = p * I - c * V + coupling;
}

extern "C" void kernel_launch(void* const* d_in, const int* in_sizes, int n_in,
                              void* d_out, int out_size, void* d_ws, size_t ws_size,
                              hipStream_t stream) {
    // input order: t, state, betas, deltas, cs, ps, ts (all float32)
    const float* state  = (const float*)d_in[1];
    const float* betas  = (const float*)d_in[2];
    const float* deltas = (const float*)d_in[3];
    const float* cs     = (const float*)d_in[4];
    const float* ps     = (const float*)d_in[5];
    const float* ts     = (const float*)d_in[6];
    float* out = (float*)d_out;

    // Workspace: [NN floats Vvec][split*NN floats partials]
    int split = 32;
    while (split > 1 &&
           (size_t)(split + 1) * NN * sizeof(float) > ws_size)
        split >>= 1;

    float* Vvec     = (float*)d_ws;
    float* partials = Vvec + NN;

    extract_V_kernel<<<NN / 256, 256, 0, stream>>>(state, Vvec);

    dim3 grid(NN / 16, split);
    matvec_partial_kernel<<<grid, 32, 0, stream>>>(ts, Vvec, partials);

    finalize_kernel<<<NN / 256, 256, 0, stream>>>(
        state, betas, deltas, cs, ps, ts, partials, out, split);
}